// ForcastConvTransformer_41102837023486
// MI455X (gfx1250) — compile-verified
//
#include <hip/hip_runtime.h>
#include <cstddef>

// ---------------------------------------------------------------------------
// Model constants (from reference): K=38, H=8, D=4, T=512, B=32, KS=5.
// Scrambled attention: 256 pseudo-batches (B*8), S=512, d=38.
// Q padded to 40 features, V padded to 48 (3 x 16 WMMA N-tiles).
// K is stored transposed: Kt[pb][40][512] so attention B-fragment loads are
// lane-contiguous.  QKV weights are pre-transposed to K-major per layer.
// ---------------------------------------------------------------------------
#define NB 32
#define NT 512
#define NK 38
#define NH 8
#define ND 4
#define NPB 256         // NB * NH pseudo-batches
#define QK_STRIDE 40
#define V_STRIDE 48
#define CH 304          // NK * NH conv channels
#define SCALE_F 2.4828239f   // 38^0.25 in fp32
#define X5_STR 201      // im2col row stride (odd => conflict-free LDS)
#define KQK 192         // padded K extent of the Q/K im2col GEMM
#define KV 40           // padded K extent of the V GEMM

typedef float v2f __attribute__((ext_vector_type(2)));
typedef float v8f __attribute__((ext_vector_type(8)));

// ---------------------------------------------------------------------------
// Kernel 1: SE gate (only middle conv tap is live for a length-1 sequence)
// + positional table; writes xe and h0.
// ---------------------------------------------------------------------------
__global__ __launch_bounds__(256) void se_pos_kernel(
    const float* __restrict__ x, const float* __restrict__ f1w,
    const float* __restrict__ f1b, const float* __restrict__ f2w,
    const float* __restrict__ f2b, const float* __restrict__ pos,
    float* __restrict__ xe, float* __restrict__ h0) {
  const int b = blockIdx.x;
  __shared__ float mean[NK];
  __shared__ float s2[2];
  __shared__ float sg[NK];
  const int j = threadIdx.x;
  if (j < NK) {
    float sm = 0.f;
    const float* xb = x + (size_t)b * NT * NK + j;
    for (int t = 0; t < NT; ++t) sm += xb[(size_t)t * NK];
    mean[j] = sm / (float)NT;
  }
  __syncthreads();
  if (j < 2) {
    float a = f1b[j];
    for (int ci = 0; ci < NK; ++ci) a += mean[ci] * f1w[(j * NK + ci) * 3 + 1];
    s2[j] = fmaxf(a, 0.f);
  }
  __syncthreads();
  if (j < NK) {
    float a = f2b[j];
    for (int c = 0; c < 2; ++c) a += s2[c] * f2w[(j * 2 + c) * 3 + 1];
    sg[j] = 1.f / (1.f + __expf(-a));
  }
  __syncthreads();
  const size_t boff = (size_t)b * NT * NK;
  for (int e = threadIdx.x; e < NT * NK; e += 256) {
    const int kk = e % NK;
    const float v = x[boff + e] * sg[kk] + sg[kk] + pos[e];
    xe[boff + e] = v;
    h0[boff + e] = v;
  }
}

// ---------------------------------------------------------------------------
// Kernel 1b (per layer): transpose projection weights to K-major with zeroed
// K-padding rows, so GEMM B-fragment loads are coalesced and unclamped.
//   qwt/kwt: [192][304]   (qw/kw are [304][190])
//   vwt:     [40][304]    (vw is [304][38])
// ---------------------------------------------------------------------------
__global__ __launch_bounds__(256) void wprep_kernel(
    const float* __restrict__ qw, const float* __restrict__ kw,
    const float* __restrict__ vw, float* __restrict__ qwt,
    float* __restrict__ kwt, float* __restrict__ vwt) {
  const int step = gridDim.x * 256;
  for (int i = blockIdx.x * 256 + threadIdx.x; i < KQK * CH; i += step) {
    const int kf = i / CH, o = i - kf * CH;
    const bool ok = kf < 190;
    qwt[i] = ok ? qw[(size_t)o * 190 + kf] : 0.f;
    kwt[i] = ok ? kw[(size_t)o * 190 + kf] : 0.f;
  }
  for (int i = blockIdx.x * 256 + threadIdx.x; i < KV * CH; i += step) {
    const int kf = i / CH, o = i - kf * CH;
    vwt[i] = (kf < NK) ? vw[(size_t)o * NK + kf] : 0.f;
  }
}

// ---------------------------------------------------------------------------
// Kernel 2 (per layer): QKV projections as im2col WMMA GEMMs.
//   Q[dt,o] = X5[dt,(ci,j)] @ qwt[(ci,j),o] + qb  (M=64, K=192, N=304)
//   K same; V[dt,o] = X[dt,ci] @ vwt[ci,o]        (M=64, K=40,  N=304)
// X5 is the causal im2col patch of xe for this pseudo-batch, staged in LDS
// with an odd row stride (conflict-free A-fragment reads).  B fragments are
// coalesced loads from the K-major weights (L2-resident).  Q/V scatter to
// the scrambled layout; K scatters transposed (Kt[pb][40][512]).
// Jobs: (proj in {Q,K,V}) x (19 N-tiles) = 57, distributed over 8 waves.
// ---------------------------------------------------------------------------
__global__ __launch_bounds__(256) void qkv_kernel(
    const float* __restrict__ xe, const float* __restrict__ qwt,
    const float* __restrict__ qb, const float* __restrict__ kwt,
    const float* __restrict__ kb, const float* __restrict__ vwt,
    float* __restrict__ Qg, float* __restrict__ Ktg, float* __restrict__ Vg) {
  const int pb = blockIdx.x;
  const int bb = pb >> 3, pp = pb & 7;
  const int base = pp << 6;
  const int wave = threadIdx.x >> 5;
  const int lane = threadIdx.x & 31;
  const int half = lane >> 4;
  const int l16 = lane & 15;
  const int koff = half ? 2 : 0;

  __shared__ float X5[64][X5_STR];   // 64 x 201 = 51456 B

  // build im2col patch: X5[dt][ci*5+j] = xe[bb, base+dt-4+j, ci]; pad cols 0
  for (int idx = threadIdx.x; idx < 64 * X5_STR; idx += 256) {
    const int dt = idx / X5_STR, r = idx - dt * X5_STR;
    float v = 0.f;
    if (r < 190) {
      const int ci = r / 5, jj = r - ci * 5;
      const int t = base + dt - 4 + jj;
      if (t >= 0) v = xe[((size_t)bb * NT + t) * NK + ci];
    }
    X5[dt][r] = v;
  }
  __syncthreads();

  for (int jb = wave; jb < 57; jb += 8) {
    const int proj = jb / 19;            // 0=Q, 1=K, 2=V
    const int nt = jb - proj * 19;
    const int n0 = nt << 4;
    const float* wsrcT = (proj == 0) ? qwt : (proj == 1) ? kwt : vwt;
    const int nks  = (proj == 2) ? 10 : 48;
    const int rmul = (proj == 2) ? 5 : 1;
    const int radd = (proj == 2) ? 4 : 0;
    const float* wlane = wsrcT + n0 + l16;   // K-major: row stride CH

    v8f c0, c1, c2, c3;
    #pragma unroll
    for (int i = 0; i < 8; ++i) { c0[i] = 0.f; c1[i] = 0.f; c2[i] = 0.f; c3[i] = 0.f; }

    for (int ks = 0; ks < nks; ++ks) {
      const int kf = (ks << 2) + koff;
      const int ra = kf * rmul + radd;
      v2f bf;
      bf.x = wlane[(size_t)kf * CH];
      bf.y = wlane[(size_t)(kf + 1) * CH];
      v2f a0, a1, a2, a3;
      a0.x = X5[l16][ra];      a0.y = X5[l16][ra + rmul];
      a1.x = X5[16 + l16][ra]; a1.y = X5[16 + l16][ra + rmul];
      a2.x = X5[32 + l16][ra]; a2.y = X5[32 + l16][ra + rmul];
      a3.x = X5[48 + l16][ra]; a3.y = X5[48 + l16][ra + rmul];
      c0 = __builtin_amdgcn_wmma_f32_16x16x4_f32(false, a0, false, bf, (short)0, c0, false, false);
      c1 = __builtin_amdgcn_wmma_f32_16x16x4_f32(false, a1, false, bf, (short)0, c1, false, false);
      c2 = __builtin_amdgcn_wmma_f32_16x16x4_f32(false, a2, false, bf, (short)0, c2, false, false);
      c3 = __builtin_amdgcn_wmma_f32_16x16x4_f32(false, a3, false, bf, (short)0, c3, false, false);
    }

    // epilogue: bias, scale, scatter to scrambled layout (K transposed)
    const float bval = (proj == 0) ? qb[n0 + l16] : (proj == 1) ? kb[n0 + l16] : 0.f;
    const int o = n0 + l16;
    const int ki = o >> 3, hi = o & 7;
    #pragma unroll
    for (int mt = 0; mt < 4; ++mt) {
      v8f cc = (mt == 0) ? c0 : (mt == 1) ? c1 : (mt == 2) ? c2 : c3;
      #pragma unroll
      for (int v = 0; v < 8; ++v) {
        const int dt = (mt << 4) + v + (half << 3);
        const int srow = (dt << 3) + hi;
        float val = cc[v] + bval;
        if (proj == 0) {
          Qg[((size_t)pb * NT + srow) * QK_STRIDE + ki] = val / SCALE_F;
        } else if (proj == 1) {
          Ktg[(size_t)pb * (QK_STRIDE * NT) + (size_t)ki * NT + srow] = val / SCALE_F;
        } else {
          Vg[((size_t)pb * NT + srow) * V_STRIDE + ki] = val;
        }
      }
    }
  }

  // zero the feature padding (ws is poisoned by the harness)
  for (int i = threadIdx.x; i < NT * 2; i += 256) {
    const int srow = i >> 1, c = NK + (i & 1);
    Qg[((size_t)pb * NT + srow) * QK_STRIDE + c] = 0.f;
  }
  for (int i = threadIdx.x; i < NT * 2; i += 256) {
    const int r = NK + (i >> 9), s = i & 511;    // rows 38,39 of Kt
    Ktg[(size_t)pb * (QK_STRIDE * NT) + (size_t)r * NT + s] = 0.f;
  }
  for (int i = threadIdx.x; i < NT * 10; i += 256) {
    const int srow = i / 10, c = NK + (i - srow * 10);
    Vg[((size_t)pb * NT + srow) * V_STRIDE + c] = 0.f;
  }
}

// ---------------------------------------------------------------------------
// Kernel 3 (per layer): fp32 WMMA flash attention per pseudo-batch.
// 8 waves; each wave owns row-tiles m = wave, wave+8, ...  Column tiles are
// processed in groups of TWO (32 keys) per online-softmax update, halving
// the shuffle/rescale overhead per key.  K is read from its transposed
// layout so every B-fragment load is lane-contiguous.  A fully-masked
// trailing tile contributes exp(-3e38 - m) = 0, so numerics stay exact.
// ---------------------------------------------------------------------------
__global__ __launch_bounds__(256) void attn_kernel(
    const float* __restrict__ Qg, const float* __restrict__ Ktg,
    const float* __restrict__ Vg, float* __restrict__ Og) {
  const int pb = blockIdx.x;
  const int bb = pb >> 3, pp = pb & 7;
  const int wave = threadIdx.x >> 5;
  const int lane = threadIdx.x & 31;
  const int half = lane >> 4;
  const int l16 = lane & 15;
  const int koff = half ? 2 : 0;

  const float* Q  = Qg + (size_t)pb * (NT * QK_STRIDE);
  const float* Kt = Ktg + (size_t)pb * (QK_STRIDE * NT);   // [40][512]
  const float* V  = Vg + (size_t)pb * (NT * V_STRIDE);

  __shared__ float stage[8][16 * 33];   // 16 rows x 32 cols, odd stride
  float* st = stage[wave];

  for (int m = wave; m < 32; m += 8) {
    const int r0 = m << 4;
    // A fragments of Q for this row tile (held across the column loop)
    v2f qa[10];
    {
      const float* qr = Q + (size_t)(r0 + l16) * QK_STRIDE + koff;
      #pragma unroll
      for (int ks = 0; ks < 10; ++ks) {
        qa[ks].x = qr[ks * 4];
        qa[ks].y = qr[ks * 4 + 1];
      }
    }
    v8f o0, o1, o2;
    float mrow[8], lrow[8];
    #pragma unroll
    for (int i = 0; i < 8; ++i) {
      o0[i] = 0.f; o1[i] = 0.f; o2[i] = 0.f;
      mrow[i] = -3.0e38f; lrow[i] = 0.f;
    }

    const int ngroups = (m + 2) >> 1;
    for (int g = 0; g < ngroups; ++g) {
      const int c0 = g << 5;           // 32 keys per group
      v8f s0, s1;
      #pragma unroll
      for (int i = 0; i < 8; ++i) { s0[i] = 0.f; s1[i] = 0.f; }
      {
        // Kt rows = features: coalesced across lanes
        const float* kb0 = Kt + (size_t)koff * NT + c0 + l16;
        #pragma unroll
        for (int ks = 0; ks < 10; ++ks) {
          const float* kr = kb0 + (size_t)(ks * 4) * NT;
          v2f bf0, bf1;
          bf0.x = kr[0];
          bf0.y = kr[NT];
          bf1.x = kr[16];
          bf1.y = kr[NT + 16];
          s0 = __builtin_amdgcn_wmma_f32_16x16x4_f32(false, qa[ks], false, bf0, (short)0, s0, false, false);
          s1 = __builtin_amdgcn_wmma_f32_16x16x4_f32(false, qa[ks], false, bf1, (short)0, s1, false, false);
        }
      }
      const int t0 = g << 1, t1 = t0 + 1;
      if (t0 == m) {  // diagonal in first tile
        #pragma unroll
        for (int v = 0; v < 8; ++v)
          if (l16 > v + (half << 3)) s0[v] = -3.0e38f;
      }
      if (t1 == m) {  // diagonal in second tile
        #pragma unroll
        for (int v = 0; v < 8; ++v)
          if (l16 > v + (half << 3)) s1[v] = -3.0e38f;
      } else if (t1 > m) {  // trailing tile fully masked
        #pragma unroll
        for (int v = 0; v < 8; ++v) s1[v] = -3.0e38f;
      }

      float sc[8];
      #pragma unroll
      for (int v = 0; v < 8; ++v) {
        float xm = fmaxf(s0[v], s1[v]);
        xm = fmaxf(xm, __shfl_xor(xm, 1, 16));
        xm = fmaxf(xm, __shfl_xor(xm, 2, 16));
        xm = fmaxf(xm, __shfl_xor(xm, 4, 16));
        xm = fmaxf(xm, __shfl_xor(xm, 8, 16));
        const float mnew = fmaxf(mrow[v], xm);
        const float scale = __expf(mrow[v] - mnew);
        const float p0 = __expf(s0[v] - mnew);
        const float p1 = __expf(s1[v] - mnew);
        s0[v] = p0;
        s1[v] = p1;
        float ps = p0 + p1;
        ps += __shfl_xor(ps, 1, 16);
        ps += __shfl_xor(ps, 2, 16);
        ps += __shfl_xor(ps, 4, 16);
        ps += __shfl_xor(ps, 8, 16);
        lrow[v] = lrow[v] * scale + ps;
        mrow[v] = mnew;
        sc[v] = scale;
      }
      #pragma unroll
      for (int v = 0; v < 8; ++v) { o0[v] *= sc[v]; o1[v] *= sc[v]; o2[v] *= sc[v]; }

      // stage both P tiles (C layout) to LDS, reload as A fragments
      #pragma unroll
      for (int v = 0; v < 8; ++v) {
        const int row = v + (half << 3);
        st[row * 33 + l16] = s0[v];
        st[row * 33 + 16 + l16] = s1[v];
      }
      __threadfence_block();

      #pragma unroll
      for (int ks = 0; ks < 8; ++ks) {
        const int kc2 = (ks << 2) + koff;
        v2f pa;
        pa.x = st[l16 * 33 + kc2];
        pa.y = st[l16 * 33 + kc2 + 1];
        const float* v0 = V + (size_t)(c0 + kc2) * V_STRIDE;
        const float* v1 = v0 + V_STRIDE;
        v2f b0, b1, b2;
        b0.x = v0[l16];      b0.y = v1[l16];
        b1.x = v0[16 + l16]; b1.y = v1[16 + l16];
        b2.x = v0[32 + l16]; b2.y = v1[32 + l16];
        o0 = __builtin_amdgcn_wmma_f32_16x16x4_f32(false, pa, false, b0, (short)0, o0, false, false);
        o1 = __builtin_amdgcn_wmma_f32_16x16x4_f32(false, pa, false, b1, (short)0, o1, false, false);
        o2 = __builtin_amdgcn_wmma_f32_16x16x4_f32(false, pa, false, b2, (short)0, o2, false, false);
      }
      __threadfence_block();
    }

    // normalize and write: Og[bb][s][pp*38 + kk]
    #pragma unroll
    for (int v = 0; v < 8; ++v) {
      const float inv = 1.0f / lrow[v];
      const int srow = r0 + v + (half << 3);
      float* orow = Og + ((size_t)bb * NT + srow) * CH + pp * NK;
      orow[l16] = o0[v] * inv;
      orow[16 + l16] = o1[v] * inv;
      if (l16 < 6) orow[32 + l16] = o2[v] * inv;
    }
  }
}

// ---------------------------------------------------------------------------
// Kernel 4 (per layer): y = O @ uw^T + ub + xe, then LayerNorm(ln1) -> ybuf.
// ---------------------------------------------------------------------------
__global__ __launch_bounds__(64) void postattn_kernel(
    const float* __restrict__ Og, const float* __restrict__ xe,
    const float* __restrict__ uw, const float* __restrict__ ub,
    const float* __restrict__ g, const float* __restrict__ be,
    float* __restrict__ ybuf) {
  const int blk = blockIdx.x;
  const int bb = blk >> 9, s = blk & 511;
  __shared__ float orow[CH];
  __shared__ float val[NK];
  __shared__ float red[2];
  const float* src = Og + ((size_t)bb * NT + s) * CH;
  for (int i = threadIdx.x; i < CH; i += 64) orow[i] = src[i];
  __syncthreads();
  const int j = threadIdx.x;
  if (j < NK) {
    float acc = ub[j];
    const float* w = uw + (size_t)j * CH;
    for (int c = 0; c < CH; ++c) acc += orow[c] * w[c];
    acc += xe[((size_t)bb * NT + s) * NK + j];
    val[j] = acc;
  }
  __syncthreads();
  if (j == 0) {
    float sm = 0.f;
    for (int c = 0; c < NK; ++c) sm += val[c];
    const float mean = sm / (float)NK;
    float v = 0.f;
    for (int c = 0; c < NK; ++c) { const float d = val[c] - mean; v += d * d; }
    red[0] = mean;
    red[1] = 1.0f / sqrtf(v / (float)NK + 1e-5f);
  }
  __syncthreads();
  if (j < NK)
    ybuf[((size_t)bb * NT + s) * NK + j] = (val[j] - red[0]) * red[1] * g[j] + be[j];
}

// ---------------------------------------------------------------------------
// Kernel 5 (per layer): FFN + residual + LayerNorm(ln2) -> xe (new layer in).
// ---------------------------------------------------------------------------
__global__ __launch_bounds__(192) void ffn_kernel(
    const float* __restrict__ ybuf, const float* __restrict__ w1,
    const float* __restrict__ b1, const float* __restrict__ w2,
    const float* __restrict__ b2, const float* __restrict__ g,
    const float* __restrict__ be, float* __restrict__ xe) {
  const int blk = blockIdx.x;
  const int bb = blk >> 9, s = blk & 511;
  __shared__ float yrow[NK];
  __shared__ float h[152];
  __shared__ float val[NK];
  __shared__ float red[2];
  const float* src = ybuf + ((size_t)bb * NT + s) * NK;
  if (threadIdx.x < NK) yrow[threadIdx.x] = src[threadIdx.x];
  __syncthreads();
  const int j = threadIdx.x;
  if (j < 152) {
    float a = b1[j];
    const float* w = w1 + (size_t)j * NK;
    for (int c = 0; c < NK; ++c) a += yrow[c] * w[c];
    h[j] = fmaxf(a, 0.f);
  }
  __syncthreads();
  if (j < NK) {
    float a = b2[j];
    const float* w = w2 + (size_t)j * 152;
    for (int c = 0; c < 152; ++c) a += h[c] * w[c];
    val[j] = a + yrow[j];
  }
  __syncthreads();
  if (j == 0) {
    float sm = 0.f;
    for (int c = 0; c < NK; ++c) sm += val[c];
    const float mean = sm / (float)NK;
    float v = 0.f;
    for (int c = 0; c < NK; ++c) { const float d = val[c] - mean; v += d * d; }
    red[0] = mean;
    red[1] = 1.0f / sqrtf(v / (float)NK + 1e-5f);
  }
  __syncthreads();
  if (j < NK)
    xe[((size_t)bb * NT + s) * NK + j] = (val[j] - red[0]) * red[1] * g[j] + be[j];
}

// ---------------------------------------------------------------------------
// Kernel 6: mu[b][m] = tomu_b[m] + sum_n (xe+h0)[b][n] * tomu_w[m][n]
// ---------------------------------------------------------------------------
__global__ __launch_bounds__(256) void mu_kernel(
    const float* __restrict__ xe, const float* __restrict__ h0,
    const float* __restrict__ tw, const float* __restrict__ tb,
    float* __restrict__ mu) {
  const int b = blockIdx.x >> 7, m = blockIdx.x & 127;
  const float* xb = xe + (size_t)b * (NT * NK);
  const float* hb = h0 + (size_t)b * (NT * NK);
  const float* w = tw + (size_t)m * (NT * NK);
  float a = 0.f;
  for (int i = threadIdx.x; i < NT * NK; i += 256) a += (xb[i] + hb[i]) * w[i];
  __shared__ float red[256];
  red[threadIdx.x] = a;
  __syncthreads();
  for (int s = 128; s > 0; s >>= 1) {
    if (threadIdx.x < s) red[threadIdx.x] += red[threadIdx.x + s];
    __syncthreads();
  }
  if (threadIdx.x == 0) mu[b * 128 + m] = red[0] + tb[m];
}

// ---------------------------------------------------------------------------
// Kernel 7: out[b][j] = tomu1_b[j] + sum_m mu[b][m] * tomu1_w[j][m]
// ---------------------------------------------------------------------------
__global__ __launch_bounds__(64) void out_kernel(
    const float* __restrict__ mu, const float* __restrict__ w,
    const float* __restrict__ bias, float* __restrict__ out) {
  const int bb = blockIdx.x;
  const int j = threadIdx.x;
  __shared__ float mrow[128];
  for (int i = j; i < 128; i += 64) mrow[i] = mu[bb * 128 + i];
  __syncthreads();
  if (j < NK) {
    float a = bias[j];
    const float* wj = w + (size_t)j * 128;
    for (int m = 0; m < 128; ++m) a += mrow[m] * wj[m];
    out[bb * NK + j] = a;
  }
}

// ---------------------------------------------------------------------------
extern "C" void kernel_launch(void* const* d_in, const int* in_sizes, int n_in,
                              void* d_out, int out_size, void* d_ws, size_t ws_size,
                              hipStream_t stream) {
  (void)in_sizes; (void)n_in; (void)out_size; (void)ws_size;
  const float* x     = (const float*)d_in[0];
  const float* f1w   = (const float*)d_in[1];
  const float* f1b   = (const float*)d_in[2];
  const float* f2w   = (const float*)d_in[3];
  const float* f2b   = (const float*)d_in[4];
  // d_in[5..9]: act_* weights -> dead code in the reference, unused.
  const float* pos   = (const float*)d_in[10];
  const float* qw    = (const float*)d_in[11];
  const float* qb    = (const float*)d_in[12];
  const float* kw    = (const float*)d_in[13];
  const float* kb    = (const float*)d_in[14];
  const float* vw    = (const float*)d_in[15];
  const float* uw    = (const float*)d_in[16];
  const float* ub    = (const float*)d_in[17];
  const float* ln1g  = (const float*)d_in[18];
  const float* ln1b  = (const float*)d_in[19];
  const float* ffw1  = (const float*)d_in[20];
  const float* ffb1  = (const float*)d_in[21];
  const float* ffw2  = (const float*)d_in[22];
  const float* ffb2  = (const float*)d_in[23];
  const float* ln2g  = (const float*)d_in[24];
  const float* ln2b  = (const float*)d_in[25];
  const float* tomuw = (const float*)d_in[26];
  const float* tomub = (const float*)d_in[27];
  const float* tm1w  = (const float*)d_in[28];
  const float* tm1b  = (const float*)d_in[29];
  float* out = (float*)d_out;

  // workspace layout (floats), ~95 MB total
  float* W    = (float*)d_ws;
  float* xe   = W;                                   // 622592
  float* h0   = xe + (size_t)NB * NT * NK;           // 622592
  float* ybuf = h0 + (size_t)NB * NT * NK;           // 622592
  float* Og   = ybuf + (size_t)NB * NT * NK;         // 4980736
  float* Qg   = Og + (size_t)NB * NT * CH;           // 5242880
  float* Ktg  = Qg + (size_t)NPB * NT * QK_STRIDE;   // 5242880
  float* Vg   = Ktg + (size_t)NPB * NT * QK_STRIDE;  // 6291456
  float* mu   = Vg + (size_t)NPB * NT * V_STRIDE;    // 4096
  float* qwt  = mu + (size_t)NB * 128;               // 58368
  float* kwt  = qwt + (size_t)KQK * CH;              // 58368
  float* vwt  = kwt + (size_t)KQK * CH;              // 12160

  se_pos_kernel<<<NB, 256, 0, stream>>>(x, f1w, f1b, f2w, f2b, pos, xe, h0);

  for (int i = 0; i < ND; ++i) {
    const size_t wqk = (size_t)i * CH * NK * 5;   // 57760
    wprep_kernel<<<64, 256, 0, stream>>>(
        qw + wqk, kw + wqk, vw + (size_t)i * CH * NK, qwt, kwt, vwt);
    qkv_kernel<<<NPB, 256, 0, stream>>>(
        xe, qwt, qb + (size_t)i * CH, kwt, kb + (size_t)i * CH, vwt,
        Qg, Ktg, Vg);
    attn_kernel<<<NPB, 256, 0, stream>>>(Qg, Ktg, Vg, Og);
    postattn_kernel<<<NB * NT, 64, 0, stream>>>(
        Og, xe, uw + (size_t)i * NK * CH, ub + (size_t)i * NK,
        ln1g + (size_t)i * NK, ln1b + (size_t)i * NK, ybuf);
    ffn_kernel<<<NB * NT, 192, 0, stream>>>(
        ybuf, ffw1 + (size_t)i * 152 * NK, ffb1 + (size_t)i * 152,
        ffw2 + (size_t)i * NK * 152, ffb2 + (size_t)i * NK,
        ln2g + (size_t)i * NK, ln2b + (size_t)i * NK, xe);
  }

  mu_kernel<<<NB * 128, 256, 0, stream>>>(xe, h0, tomuw, tomub, mu);
  out_kernel<<<NB, 64, 0, stream>>>(mu, tm1w, tm1b, out);
}